// WaveEqnSol_35390530519880
// MI455X (gfx1250) — compile-verified
//
#include <hip/hip_runtime.h>
#include <math.h>
#include <stdint.h>

// CDNA5 (gfx1250) wave32 WMMA types
typedef __attribute__((ext_vector_type(2))) float v2f;
typedef __attribute__((ext_vector_type(8))) float v8f;
typedef __attribute__((ext_vector_type(4))) unsigned u32x4;
typedef __attribute__((ext_vector_type(4))) int i32x4;
typedef __attribute__((ext_vector_type(8))) int i32x8;

#define EPS 1e-5f

// Block: 128 threads = 4 waves arranged 2(M) x 2(N); wave tile 32x32 (2x2 WMMA).
// Block tile 64x64, K-tile 32, double-buffered LDS fed by the Tensor Data Mover.
#define BM 64
#define BN 64
#define TK 32
#define S1 72   // k-major LDS row stride: 64 data + 8 pad dwords (TDM pad 5/7); 2*72%64=16 -> disjoint lane-half banks
#define SA 34   // pass-2 A k-minor row stride: 32 data + 2 pad dwords (TDM pad 4/1); even -> aligned b64 frags

static __device__ __forceinline__ v8f wmma4(v2f a, v2f b, v8f c) {
  return __builtin_amdgcn_wmma_f32_16x16x4_f32(false, a, false, b, (short)0, c,
                                               false, false);
}

// ---------------------------------------------------------------------------
// TDM 2D tile load: global (row-major, element stride0 per row) -> LDS with
// hardware row padding. D# packing per CDNA5 ISA 8.3/8.4.
//   group0: count=1 | lds_addr | global_addr[56:0] | type=2
//   group1: data_size=4B, pad_enable, pad_interval, pad_amount,
//           tensor_dim0/1, tile_dim0/1 (tile_dim2=0), tensor_dim0_stride
// ---------------------------------------------------------------------------
static __device__ __forceinline__ void tdm_load_2d(
    uint32_t lds_addr, const void* gp, uint32_t tdim0, uint32_t tdim1,
    uint32_t tile0, uint32_t tile1, uint64_t stride0_elems,
    uint32_t pad_iv, uint32_t pad_amt) {
  uint64_t ga = (uint64_t)(uintptr_t)gp;
  u32x4 g0;
  g0.x = 1u;                                   // count=1 (valid descriptor)
  g0.y = lds_addr;                             // LDS byte address
  g0.z = (uint32_t)ga;                         // global_addr[31:0]
  g0.w = (uint32_t)((ga >> 32) & 0x01FFFFFFu)  // global_addr[56:32]
         | 0x80000000u;                        // type=2 ("image")
  i32x8 g1;
  g1[0] = (int)((2u << 16)                     // data_size: 4 bytes
               | (1u << 20)                    // pad_enable
               | (pad_iv << 22) | (pad_amt << 25));
  g1[1] = (int)((tdim0 & 0xFFFFu) << 16);                       // tensor_dim0[15:0]
  g1[2] = (int)((tdim0 >> 16) | ((tdim1 & 0xFFFFu) << 16));     // dim0[31:16]|dim1[15:0]
  g1[3] = (int)((tdim1 >> 16) | (tile0 << 16));                 // dim1[31:16]|tile_dim0
  g1[4] = (int)tile1;                                           // tile_dim1 (tile_dim2=0)
  g1[5] = (int)(uint32_t)stride0_elems;                         // tensor_dim0_stride[31:0]
  g1[6] = (int)(uint32_t)(stride0_elems >> 32);                 // stride[47:32] (stride1=0)
  g1[7] = 0;
  i32x4 z4 = {0, 0, 0, 0};
#if __clang_major__ >= 23
  i32x8 z8 = {0, 0, 0, 0, 0, 0, 0, 0};
  __builtin_amdgcn_tensor_load_to_lds(g0, g1, z4, z4, z8, 0);
#else
  __builtin_amdgcn_tensor_load_to_lds(g0, g1, z4, z4, 0);
#endif
}

static __device__ __forceinline__ uint32_t lds_off(const void* p) {
  return (uint32_t)(uintptr_t)p;   // generic LDS pointer: low 32 bits = LDS offset
}

// ---------------------------------------------------------------------------
// Spectral weights: c = cos(t*sqrt(max(lam,0))); p2 = sin(t*s)/s (limit t).
// ---------------------------------------------------------------------------
__global__ void spectrum_kernel(const float* __restrict__ lam,
                                const float* __restrict__ t_ptr,
                                float* __restrict__ c, float* __restrict__ p2,
                                int n) {
  int i = blockIdx.x * blockDim.x + threadIdx.x;
  if (i >= n) return;
  float t = *t_ptr;
  float s = sqrtf(fmaxf(lam[i], 0.0f));
  c[i] = cosf(t * s);
  bool small = fabsf(s) < EPS;
  float ss = small ? 1.0f : s;
  p2[i] = small ? t : (sinf(t * ss) / ss);
}

// ---------------------------------------------------------------------------
// Pass 1 (fused): W[i,f] = c[i]*sum_j V[j,i]*X[j,f] + p2[i]*sum_j V[j,i]*Y[j,f]
// TDM stages k-major panels: As/Bxs/Bys[buf][k][S1].
// ---------------------------------------------------------------------------
__global__ __launch_bounds__(128) void gemm_vt_fused(
    const float* __restrict__ V, const float* __restrict__ X,
    const float* __restrict__ Y, const float* __restrict__ c,
    const float* __restrict__ p2, float* __restrict__ W, int n, int F) {
  __shared__ float As [2][TK][S1];
  __shared__ float Bxs[2][TK][S1];
  __shared__ float Bys[2][TK][S1];

  const int tid  = threadIdx.x;
  const int lane = tid & 31;
  const int wave = tid >> 5;
  const int wm   = wave & 1;
  const int wn   = wave >> 1;
  const int i0   = blockIdx.x * BM;
  const int f0   = blockIdx.y * BN;
  const int ml   = lane & 15;
  const int kq   = (lane >> 4) << 1;   // lane-half k offset (ISA 7.12.2 f32 layout)

  v8f accx[2][2] = {};
  v8f accy[2][2] = {};

  const int nt = n / TK;

  if (wave == 0) {  // TDM ignores EXEC: confine issue to one wave by branch
    tdm_load_2d(lds_off(&As [0][0][0]), &V[i0],          n, n, BM, TK, (uint64_t)n, 5, 7);
    tdm_load_2d(lds_off(&Bxs[0][0][0]), &X[f0],          F, n, BN, TK, (uint64_t)F, 5, 7);
    tdm_load_2d(lds_off(&Bys[0][0][0]), &Y[f0],          F, n, BN, TK, (uint64_t)F, 5, 7);
  }

  for (int kt = 0; kt < nt; ++kt) {
    const int buf = kt & 1;
    const int nb  = buf ^ 1;
    __syncthreads();                   // all waves done reading buffer `nb`
    if (wave == 0) {
      if (kt + 1 < nt) {
        const size_t k0n = (size_t)(kt + 1) * TK;
        tdm_load_2d(lds_off(&As [nb][0][0]), &V[k0n * n + i0], n, n, BM, TK, (uint64_t)n, 5, 7);
        tdm_load_2d(lds_off(&Bxs[nb][0][0]), &X[k0n * F + f0], F, n, BN, TK, (uint64_t)F, 5, 7);
        tdm_load_2d(lds_off(&Bys[nb][0][0]), &Y[k0n * F + f0], F, n, BN, TK, (uint64_t)F, 5, 7);
        __builtin_amdgcn_s_wait_tensorcnt(3);   // tile kt's 3 DMAs complete
      } else {
        __builtin_amdgcn_s_wait_tensorcnt(0);
      }
    }
    __syncthreads();                   // tile kt visible to all waves

    const int ra = wm * 32 + ml;
    const int rb = wn * 32 + ml;
#pragma unroll
    for (int kk = 0; kk < TK / 4; ++kk) {
      const int kb = kk * 4 + kq;
      v2f a0, a1, b0, b1, d0, d1;
      a0.x = As [buf][kb][ra];      a0.y = As [buf][kb + 1][ra];
      a1.x = As [buf][kb][ra + 16]; a1.y = As [buf][kb + 1][ra + 16];
      b0.x = Bxs[buf][kb][rb];      b0.y = Bxs[buf][kb + 1][rb];
      b1.x = Bxs[buf][kb][rb + 16]; b1.y = Bxs[buf][kb + 1][rb + 16];
      d0.x = Bys[buf][kb][rb];      d0.y = Bys[buf][kb + 1][rb];
      d1.x = Bys[buf][kb][rb + 16]; d1.y = Bys[buf][kb + 1][rb + 16];
      accx[0][0] = wmma4(a0, b0, accx[0][0]);
      accx[0][1] = wmma4(a0, b1, accx[0][1]);
      accx[1][0] = wmma4(a1, b0, accx[1][0]);
      accx[1][1] = wmma4(a1, b1, accx[1][1]);
      accy[0][0] = wmma4(a0, d0, accy[0][0]);
      accy[0][1] = wmma4(a0, d1, accy[0][1]);
      accy[1][0] = wmma4(a1, d0, accy[1][0]);
      accy[1][1] = wmma4(a1, d1, accy[1][1]);
    }
  }

  // Epilogue: scale by spectra. C/D layout: VGPR r -> M = r + 8*(lane>>4), N = lane%16.
  const int lh = (lane >> 4) << 3;
#pragma unroll
  for (int mi = 0; mi < 2; ++mi) {
#pragma unroll
    for (int ni = 0; ni < 2; ++ni) {
      const int col  = f0 + wn * 32 + ni * 16 + ml;
      const int rowb = i0 + wm * 32 + mi * 16 + lh;
#pragma unroll
      for (int r = 0; r < 8; ++r) {
        int row = rowb + r;
        W[row * F + col] = c[row] * accx[mi][ni][r] + p2[row] * accy[mi][ni][r];
      }
    }
  }
}

// ---------------------------------------------------------------------------
// Pass 2: out[i,f] = sum_j V[i,j] * W[j,f]
// A panel is k-contiguous in memory -> TDM lands it k-minor Aps[buf][m][SA]
// (keeps b64 fragment loads); B panel k-major like pass 1.
// ---------------------------------------------------------------------------
__global__ __launch_bounds__(128) void gemm_v(
    const float* __restrict__ V, const float* __restrict__ W,
    float* __restrict__ out, int n, int F) {
  __shared__ float Aps[2][BM][SA];
  __shared__ float Bs [2][TK][S1];

  const int tid  = threadIdx.x;
  const int lane = tid & 31;
  const int wave = tid >> 5;
  const int wm   = wave & 1;
  const int wn   = wave >> 1;
  const int i0   = blockIdx.x * BM;
  const int f0   = blockIdx.y * BN;
  const int ml   = lane & 15;
  const int kq   = (lane >> 4) << 1;

  v8f acc[2][2] = {};

  const int nt = n / TK;

  if (wave == 0) {
    tdm_load_2d(lds_off(&Aps[0][0][0]), &V[(size_t)i0 * n], n, n, TK, BM, (uint64_t)n, 4, 1);
    tdm_load_2d(lds_off(&Bs [0][0][0]), &W[f0],             F, n, BN, TK, (uint64_t)F, 5, 7);
  }

  for (int kt = 0; kt < nt; ++kt) {
    const int buf = kt & 1;
    const int nb  = buf ^ 1;
    __syncthreads();
    if (wave == 0) {
      if (kt + 1 < nt) {
        const size_t k0n = (size_t)(kt + 1) * TK;
        tdm_load_2d(lds_off(&Aps[nb][0][0]), &V[(size_t)i0 * n + k0n], n, n, TK, BM, (uint64_t)n, 4, 1);
        tdm_load_2d(lds_off(&Bs [nb][0][0]), &W[k0n * F + f0],         F, n, BN, TK, (uint64_t)F, 5, 7);
        __builtin_amdgcn_s_wait_tensorcnt(2);
      } else {
        __builtin_amdgcn_s_wait_tensorcnt(0);
      }
    }
    __syncthreads();

    const int ra = wm * 32 + ml;
    const int rb = wn * 32 + ml;
#pragma unroll
    for (int kk = 0; kk < TK / 4; ++kk) {
      const int kb = kk * 4 + kq;
      v2f a0, a1, b0, b1;
      a0.x = Aps[buf][ra     ][kb]; a0.y = Aps[buf][ra     ][kb + 1];
      a1.x = Aps[buf][ra + 16][kb]; a1.y = Aps[buf][ra + 16][kb + 1];
      b0.x = Bs [buf][kb][rb];      b0.y = Bs [buf][kb + 1][rb];
      b1.x = Bs [buf][kb][rb + 16]; b1.y = Bs [buf][kb + 1][rb + 16];
      acc[0][0] = wmma4(a0, b0, acc[0][0]);
      acc[0][1] = wmma4(a0, b1, acc[0][1]);
      acc[1][0] = wmma4(a1, b0, acc[1][0]);
      acc[1][1] = wmma4(a1, b1, acc[1][1]);
    }
  }

  const int lh = (lane >> 4) << 3;
#pragma unroll
  for (int mi = 0; mi < 2; ++mi) {
#pragma unroll
    for (int ni = 0; ni < 2; ++ni) {
      const int col  = f0 + wn * 32 + ni * 16 + ml;
      const int rowb = i0 + wm * 32 + mi * 16 + lh;
#pragma unroll
      for (int r = 0; r < 8; ++r) {
        out[(rowb + r) * F + col] = acc[mi][ni][r];
      }
    }
  }
}

// ---------------------------------------------------------------------------
// Launch: inputs in dict order {x, y, t, lam, V}
// ---------------------------------------------------------------------------
extern "C" void kernel_launch(void* const* d_in, const int* in_sizes, int n_in,
                              void* d_out, int out_size, void* d_ws, size_t ws_size,
                              hipStream_t stream) {
  const float* x   = (const float*)d_in[0];
  const float* y   = (const float*)d_in[1];
  const float* t   = (const float*)d_in[2];
  const float* lam = (const float*)d_in[3];
  const float* V   = (const float*)d_in[4];
  float* out = (float*)d_out;

  const int n = in_sizes[3];          // 4096
  const int F = in_sizes[0] / n;      // 256

  float* ws = (float*)d_ws;
  float* c  = ws;                     // [n]
  float* p2 = ws + n;                 // [n]
  float* W  = ws + 2 * n;             // [n*F]

  spectrum_kernel<<<(n + 255) / 256, 256, 0, stream>>>(lam, t, c, p2, n);

  dim3 grid(n / BM, F / BN);          // 64 x 4 blocks
  gemm_vt_fused<<<grid, 128, 0, stream>>>(V, x, y, c, p2, W, n, F);
  gemm_v<<<grid, 128, 0, stream>>>(V, W, out, n, F);
}